// DiffnetPlusMod_3745211482781
// MI455X (gfx1250) — compile-verified
//
#include <hip/hip_runtime.h>
#include <hip/hip_bf16.h>
#include <math.h>

#define NU 100000
#define NI 50000
#define RD 300
#define DD 64
#define NE_UCI 1200000
#define NE_UN  600000
#define NE_ICU 1200000
#define NE_IN  600000
#define NBATCH 8192
#define EPSN 1e-5f

typedef __attribute__((ext_vector_type(2))) float v2f;
typedef __attribute__((ext_vector_type(8))) float v8f;

// ---------------- utility kernels ----------------

__global__ void zero_f32(float* __restrict__ p, long n) {
    long i = (long)blockIdx.x * blockDim.x + threadIdx.x;
    long st = (long)gridDim.x * blockDim.x;
    for (; i < n; i += st) p[i] = 0.f;
}

__global__ void reduce_stats_kernel(const float* __restrict__ x, long n,
                                    float* __restrict__ out2) {
    __shared__ float ss[256], sq[256];
    float s = 0.f, q = 0.f;
    long i = (long)blockIdx.x * blockDim.x + threadIdx.x;
    long st = (long)gridDim.x * blockDim.x;
    for (; i < n; i += st) { float v = x[i]; s += v; q += v * v; }
    ss[threadIdx.x] = s; sq[threadIdx.x] = q; __syncthreads();
    for (int k = 128; k > 0; k >>= 1) {
        if (threadIdx.x < k) { ss[threadIdx.x] += ss[threadIdx.x + k];
                               sq[threadIdx.x] += sq[threadIdx.x + k]; }
        __syncthreads();
    }
    if (threadIdx.x == 0) { atomicAdd(&out2[0], ss[0]); atomicAdd(&out2[1], sq[0]); }
}

// ---------------- WMMA fp32 GEMM: Out[M,64] = act(norm(A[M,K]) @ B[K,64] + bias) ----------------
// norm uses (sum,sumsq) in stats_in; optionally accumulates (sum,sumsq) of output into stats_out.
// One wave computes a 16x16 tile; 4 waves/block cover N=64. EXEC all-ones (M%16==0, K%4==0).

template<int K, bool SIG, bool ACC>
__global__ __launch_bounds__(128) void gemm_wmma_f32(
    const float* __restrict__ A, const float* __restrict__ B,
    const float* __restrict__ bias, const float* __restrict__ stats_in,
    float invN, float* __restrict__ Out, float* __restrict__ stats_out) {
    const int wave = threadIdx.x >> 5;
    const int lane = threadIdx.x & 31;
    const int lo = lane & 15;
    const int hi = lane >> 4;
    const int row0 = blockIdx.x * 16;
    const int col0 = wave * 16;

    const float mean = stats_in[0] * invN;
    const float var  = stats_in[1] * invN - mean * mean;
    const float rstd = rsqrtf(var + EPSN);

    v8f acc = {};
    const float* Arow = A + (long)(row0 + lo) * K;
    #pragma unroll 5
    for (int k = 0; k < K; k += 4) {
        const int ka = k + 2 * hi;                       // A/B f32 layout: K = 2*(lane/16)+vgpr
        v2f a, b;
        a.x = (Arow[ka]     - mean) * rstd;
        a.y = (Arow[ka + 1] - mean) * rstd;
        b.x = B[(long)ka * DD + col0 + lo];
        b.y = B[(long)(ka + 1) * DD + col0 + lo];
        acc = __builtin_amdgcn_wmma_f32_16x16x4_f32(
            /*neg_a=*/false, a, /*neg_b=*/false, b,
            /*c_mod=*/(short)0, acc, /*reuse_a=*/false, /*reuse_b=*/false);
    }

    const float bc = bias[col0 + lo];
    float lsum = 0.f, lsq = 0.f;
    #pragma unroll
    for (int r = 0; r < 8; ++r) {                        // C/D layout: M = r + 8*(lane/16)
        const int row = row0 + r + 8 * hi;
        float v = acc[r] + bc;
        if (SIG) v = 1.f / (1.f + expf(-v));
        Out[(long)row * DD + col0 + lo] = v;
        if (ACC) { lsum += v; lsq += v * v; }
    }
    if (ACC) {
        __shared__ float ss[128], sq[128];
        ss[threadIdx.x] = lsum; sq[threadIdx.x] = lsq; __syncthreads();
        for (int k = 64; k > 0; k >>= 1) {
            if (threadIdx.x < k) { ss[threadIdx.x] += ss[threadIdx.x + k];
                                   sq[threadIdx.x] += sq[threadIdx.x + k]; }
            __syncthreads();
        }
        if (threadIdx.x == 0) { atomicAdd(&stats_out[0], ss[0]); atomicAdd(&stats_out[1], sq[0]); }
    }
}

// ---------------- edge attention + row sums ----------------
// seg-softmax(exp(sigmoid(x*w+b))) == exp(v)/segsum(exp(v)) since v in (1,e): no max pass needed.

__global__ void edge_att_kernel(const float* __restrict__ vals, const int* __restrict__ rows,
                                const float* __restrict__ wp, const float* __restrict__ bp,
                                float* __restrict__ t, float* __restrict__ rowsum, int E) {
    int e = blockIdx.x * blockDim.x + threadIdx.x;
    if (e >= E) return;
    const float w = wp[0], b = bp[0];
    const float sg = 1.f / (1.f + expf(-(vals[e] * w + b)));
    const float te = expf(expf(sg));
    t[e] = te;
    atomicAdd(&rowsum[rows[e]], te);
}

// ---------------- SPMM scatter: out[row] += (t/rowsum[row]) * src[col], 16 lanes/edge ----------------

__global__ void spmm_kernel(const float* __restrict__ t, const float* __restrict__ rowsum,
                            const int* __restrict__ rows, const int* __restrict__ cols,
                            const float* __restrict__ src, float* __restrict__ out, int E) {
    long gid = (long)blockIdx.x * blockDim.x + threadIdx.x;
    int e = (int)(gid >> 4);
    int q = (int)(gid & 15);
    if (e >= E) return;
    const int r = rows[e], c = cols[e];
    const float a = t[e] / rowsum[r];
    const float4 s = *reinterpret_cast<const float4*>(src + (long)c * DD + q * 4);
    float* o = out + (long)r * DD + q * 4;
    atomicAdd(o + 0, a * s.x); atomicAdd(o + 1, a * s.y);
    atomicAdd(o + 2, a * s.z); atomicAdd(o + 3, a * s.w);
}

// ---------------- graph attention + blend: one wave per node ----------------

__global__ __launch_bounds__(256) void combine_kernel(
    const float* __restrict__ cur, const float* __restrict__ nb, const float* __restrict__ lk,
    const float* __restrict__ w1a, const float* __restrict__ b1a,
    const float* __restrict__ w2a, const float* __restrict__ b2a, float ca,
    const float* __restrict__ w1b, const float* __restrict__ b1b,
    const float* __restrict__ w2b, const float* __restrict__ b2b, float cb,
    float* __restrict__ outp, int N) {
    const int node = blockIdx.x * 8 + (threadIdx.x >> 5);
    const int lane = threadIdx.x & 31;
    if (node >= N) return;
    const float* cu = cur + (long)node * DD;
    const float* nbp = nb + (long)node * DD;
    const float* lkp = lk + (long)node * DD;
    float dA = 0.f, dB = 0.f;
    #pragma unroll
    for (int d = lane; d < DD; d += 32) {
        const float c = cu[d], n = nbp[d], l = lkp[d];
        dA += (c + n) * w1a[d];
        dB += c * w1b[d] + l * w1b[DD + d];
    }
    for (int m = 16; m > 0; m >>= 1) { dA += __shfl_xor(dA, m, 32); dB += __shfl_xor(dB, m, 32); }
    float hA = tanhf(dA + b1a[0]) * w2a[0] + b2a[0];
    hA = hA >= 0.f ? hA : 0.2f * hA;
    const float gA = expf(hA) + ca;
    float hB = tanhf(dB + b1b[0]) * w2b[0] + b2b[0];
    hB = hB >= 0.f ? hB : 0.2f * hB;
    const float gB = expf(hB) + cb;
    const float inv = 1.f / (gA + gB);
    const float fa = gA * inv, fb = gB * inv;
    float* op = outp + (long)node * DD;
    #pragma unroll
    for (int d = lane; d < DD; d += 32)
        op[d] = 0.5f * cu[d] + 0.5f * (fa * nbp[d] + fb * lkp[d]);
}

// ---------------- final gather + dot over the 3x64 concat: one wave per batch element ----------------

__global__ __launch_bounds__(256) void final_dot_kernel(
    const float* __restrict__ u0, const float* __restrict__ u1, const float* __restrict__ u2,
    const float* __restrict__ i0, const float* __restrict__ i1, const float* __restrict__ i2,
    const int* __restrict__ ui, const int* __restrict__ ii, float* __restrict__ out, int B) {
    const int b = blockIdx.x * 8 + (threadIdx.x >> 5);
    const int lane = threadIdx.x & 31;
    if (b >= B) return;
    const long ur = (long)ui[b] * DD, ir = (long)ii[b] * DD;
    float s = 0.f;
    #pragma unroll
    for (int d = lane; d < DD; d += 32)
        s += u0[ur + d] * i0[ir + d] + u1[ur + d] * i1[ir + d] + u2[ur + d] * i2[ir + d];
    for (int m = 16; m > 0; m >>= 1) s += __shfl_xor(s, m, 32);
    if (lane == 0) out[b] = s;
}

// ---------------- host-side orchestration ----------------

// input indices (params dict flattened in insertion order, then edge/batch arrays)
enum {
    IN_UR = 0, IN_IR, IN_WUR, IN_BUR, IN_WIR, IN_BIR, IN_WUF, IN_BUF, IN_WIF, IN_BIF,
    IN_UCI_V, IN_UN_V, IN_ICU_V, IN_IN_V,
    IN_UCI_AW, IN_UCI_AB, IN_UN_AW, IN_UN_AB, IN_ICU_AW, IN_ICU_AB, IN_IN_AW, IN_IN_AB,
    IN_UCI_G1W, IN_UCI_G1B, IN_UCI_G2W, IN_UCI_G2B,
    IN_UN_G1W, IN_UN_G1B, IN_UN_G2W, IN_UN_G2B,
    IN_ICU_G1W, IN_ICU_G1B, IN_ICU_G2W, IN_ICU_G2B,
    IN_IN_G1W, IN_IN_G1B, IN_IN_G2W, IN_IN_G2B,
    IN_UCI_R, IN_UCI_C, IN_UN_R, IN_UN_C, IN_ICU_R, IN_ICU_C, IN_IN_R, IN_IN_C,
    IN_USER, IN_ITEM
};

extern "C" void kernel_launch(void* const* d_in, const int* in_sizes, int n_in,
                              void* d_out, int out_size, void* d_ws, size_t ws_size,
                              hipStream_t stream) {
    (void)in_sizes; (void)n_in; (void)out_size; (void)ws_size;
    const float* F[48];
    const int*   I[48];
    for (int k = 0; k < 48; ++k) { F[k] = (const float*)d_in[k]; I[k] = (const int*)d_in[k]; }

    // workspace layout (floats)
    float* w = (float*)d_ws;
    size_t o = 0;
    float* stats = w + o; o += 16;                  // [0,1]=u raw, [2,3]=u hidden, [4,5]=i raw, [6,7]=i hidden
    float* u0 = w + o; o += (size_t)NU * DD;
    float* u1 = w + o; o += (size_t)NU * DD;
    float* u2 = w + o; o += (size_t)NU * DD;
    float* i0 = w + o; o += (size_t)NI * DD;
    float* i1 = w + o; o += (size_t)NI * DD;
    float* i2 = w + o; o += (size_t)NI * DD;
    float* h   = w + o; o += (size_t)NU * DD;       // stage-1 hidden (reused for items)
    float* ufi = w + o; o += (size_t)NU * DD;       // u_from_items
    float* ufl = w + o; o += (size_t)NU * DD;       // u_from_links
    float* ifu = w + o; o += (size_t)NI * DD;       // i_from_users
    float* ifl = w + o; o += (size_t)NI * DD;       // i_from_links
    float* te  = w + o; o += (size_t)NE_UCI;        // per-edge attention (max E)
    float* rowsum = w + o; o += (size_t)NU;         // max(NU,NI)

    const dim3 B256(256);

    // ---- stats for the two input norms ----
    zero_f32<<<1, 32, 0, stream>>>(stats, 16);
    reduce_stats_kernel<<<2048, 256, 0, stream>>>(F[IN_UR], (long)NU * RD, stats + 0);
    reduce_stats_kernel<<<2048, 256, 0, stream>>>(F[IN_IR], (long)NI * RD, stats + 4);

    // ---- dense fusion pipeline (WMMA f32) ----
    gemm_wmma_f32<RD, true, true><<<NU / 16, 128, 0, stream>>>(
        F[IN_UR], F[IN_WUR], F[IN_BUR], stats + 0, 1.f / ((float)NU * RD), h, stats + 2);
    gemm_wmma_f32<DD, false, false><<<NU / 16, 128, 0, stream>>>(
        h, F[IN_WUF], F[IN_BUF], stats + 2, 1.f / ((float)NU * DD), u0, nullptr);
    gemm_wmma_f32<RD, true, true><<<NI / 16, 128, 0, stream>>>(
        F[IN_IR], F[IN_WIR], F[IN_BIR], stats + 4, 1.f / ((float)NI * RD), h, stats + 6);
    gemm_wmma_f32<DD, false, false><<<NI / 16, 128, 0, stream>>>(
        h, F[IN_WIF], F[IN_BIF], stats + 6, 1.f / ((float)NI * DD), i0, nullptr);

    // ---- 2 GCN layers ----
    float* ulist[3] = { u0, u1, u2 };
    float* ilist[3] = { i0, i1, i2 };
    for (int layer = 0; layer < 2; ++layer) {
        float* cu = ulist[layer]; float* nu = ulist[layer + 1];
        float* ci = ilist[layer]; float* ni = ilist[layer + 1];

        // user <- items (uci)
        zero_f32<<<256, 256, 0, stream>>>(rowsum, NU);
        edge_att_kernel<<<(NE_UCI + 255) / 256, B256, 0, stream>>>(
            F[IN_UCI_V], I[IN_UCI_R], F[IN_UCI_AW], F[IN_UCI_AB], te, rowsum, NE_UCI);
        zero_f32<<<2048, 256, 0, stream>>>(ufi, (long)NU * DD);
        spmm_kernel<<<(int)(((long)NE_UCI * 16 + 255) / 256), B256, 0, stream>>>(
            te, rowsum, I[IN_UCI_R], I[IN_UCI_C], ci, ufi, NE_UCI);
        // user <- users (un)
        zero_f32<<<256, 256, 0, stream>>>(rowsum, NU);
        edge_att_kernel<<<(NE_UN + 255) / 256, B256, 0, stream>>>(
            F[IN_UN_V], I[IN_UN_R], F[IN_UN_AW], F[IN_UN_AB], te, rowsum, NE_UN);
        zero_f32<<<2048, 256, 0, stream>>>(ufl, (long)NU * DD);
        spmm_kernel<<<(int)(((long)NE_UN * 16 + 255) / 256), B256, 0, stream>>>(
            te, rowsum, I[IN_UN_R], I[IN_UN_C], cu, ufl, NE_UN);
        // blend user
        combine_kernel<<<(NU + 7) / 8, B256, 0, stream>>>(
            cu, ufi, ufl,
            F[IN_UCI_G1W], F[IN_UCI_G1B], F[IN_UCI_G2W], F[IN_UCI_G2B], 0.7f,
            F[IN_UN_G1W],  F[IN_UN_G1B],  F[IN_UN_G2W],  F[IN_UN_G2B],  0.3f,
            nu, NU);

        // item <- items (in)
        zero_f32<<<256, 256, 0, stream>>>(rowsum, NI);
        edge_att_kernel<<<(NE_IN + 255) / 256, B256, 0, stream>>>(
            F[IN_IN_V], I[IN_IN_R], F[IN_IN_AW], F[IN_IN_AB], te, rowsum, NE_IN);
        zero_f32<<<2048, 256, 0, stream>>>(ifl, (long)NI * DD);
        spmm_kernel<<<(int)(((long)NE_IN * 16 + 255) / 256), B256, 0, stream>>>(
            te, rowsum, I[IN_IN_R], I[IN_IN_C], ci, ifl, NE_IN);
        // item <- users (icu)
        zero_f32<<<256, 256, 0, stream>>>(rowsum, NI);
        edge_att_kernel<<<(NE_ICU + 255) / 256, B256, 0, stream>>>(
            F[IN_ICU_V], I[IN_ICU_R], F[IN_ICU_AW], F[IN_ICU_AB], te, rowsum, NE_ICU);
        zero_f32<<<2048, 256, 0, stream>>>(ifu, (long)NI * DD);
        spmm_kernel<<<(int)(((long)NE_ICU * 16 + 255) / 256), B256, 0, stream>>>(
            te, rowsum, I[IN_ICU_R], I[IN_ICU_C], cu, ifu, NE_ICU);
        // blend item
        combine_kernel<<<(NI + 7) / 8, B256, 0, stream>>>(
            ci, ifu, ifl,
            F[IN_ICU_G1W], F[IN_ICU_G1B], F[IN_ICU_G2W], F[IN_ICU_G2B], 0.5f,
            F[IN_IN_G1W],  F[IN_IN_G1B],  F[IN_IN_G2W],  F[IN_IN_G2B],  0.5f,
            ni, NI);
    }

    // ---- final gather + dot ----
    final_dot_kernel<<<(NBATCH + 7) / 8, B256, 0, stream>>>(
        u0, u1, u2, i0, i1, i2, I[IN_USER], I[IN_ITEM], (float*)d_out, NBATCH);
}